// GCNII_72645076845143
// MI455X (gfx1250) — compile-verified
//
#include <hip/hip_runtime.h>
#include <math.h>

#define NNODES 100000
#define NFEAT  128
#define NHID   128
#define NCLASS 64
#define NEDGES 600000
#define ALPHA  0.1f
#define BETA   0.5f

#define APAD 132   // A-tile LDS row stride (floats): bank stride 4 -> conflict-free b64 reads
#define WPAD 136   // W LDS row stride (floats): halves 16 banks apart -> conflict-free b32 reads
#define OPAD 72    // W_out LDS row stride (floats)

typedef float v2f __attribute__((ext_vector_type(2)));
typedef float v8f __attribute__((ext_vector_type(8)));

// ---------------------------------------------------------------------------
// One wave32 computes a 16x16 f32 tile via V_WMMA_F32_16X16X4_F32.
// A frag (ISA 7.12.2, 32-bit A 16x4): m = lane&15, k base = (lane>=16)?2:0.
// B frag mirrored over N. C/D: VGPR v -> M = v + 8*(lane>=16), N = lane&15.
// All operands served from padded LDS (conflict-free banking, see strides).
// ---------------------------------------------------------------------------

// h = relu(x @ W_in + b); also writes h0 = h
__global__ __launch_bounds__(256)
void k_lin_in(const float* __restrict__ x, const float* __restrict__ W,
              const float* __restrict__ bias, float* __restrict__ h,
              float* __restrict__ h0)
{
    __shared__ float As[16 * APAD];        // 8.4 KB
    __shared__ float Ws[64 * WPAD];        // 34.8 KB (one K-half of W)
    const int tid  = threadIdx.x;
    const int wave = tid >> 5;
    const int lane = tid & 31;
    const int half = lane >> 4;
    const int idx  = lane & 15;
    const int rowBase = blockIdx.x * 16;
    const int colBase = wave * 16;
    const int col = colBase + idx;

    // stage 16x128 A tile (512 float4, 2 per thread), padded rows
#pragma unroll
    for (int j = tid; j < 512; j += 256) {
        const int r = j >> 5, c = (j & 31) << 2;
        *(float4*)&As[r * APAD + c] =
            *(const float4*)&x[(size_t)(rowBase + r) * NFEAT + c];
    }

    v8f acc = {};
    for (int hk = 0; hk < 2; ++hk) {
        const int ks = hk * 64;
        __syncthreads();
        // stage K-half of W: 64x128 (2048 float4, 8 per thread), padded rows
#pragma unroll
        for (int j = tid; j < 2048; j += 256) {
            const int r = j >> 5, c = (j & 31) << 2;
            *(float4*)&Ws[r * WPAD + c] =
                *(const float4*)&W[(size_t)(ks + r) * NHID + c];
        }
        __syncthreads();
#pragma unroll
        for (int k = 0; k < 64; k += 4) {
            v2f a, b;
            a.x = As[idx * APAD + ks + k + half * 2 + 0];
            a.y = As[idx * APAD + ks + k + half * 2 + 1];
            b.x = Ws[(k + half * 2 + 0) * WPAD + col];
            b.y = Ws[(k + half * 2 + 1) * WPAD + col];
            acc = __builtin_amdgcn_wmma_f32_16x16x4_f32(false, a, false, b,
                                                        (short)0, acc, false, false);
        }
    }

    const float bv = bias[col];
#pragma unroll
    for (int v = 0; v < 8; ++v) {
        const int row = rowBase + v + half * 8;
        float val = acc[v] + bv;
        val = val > 0.f ? val : 0.f;
        h [(size_t)row * NHID + col] = val;
        h0[(size_t)row * NHID + col] = val;
    }
}

// support = (1-alpha)*h + alpha*h0 ; agg = 0   (float4-vectorized)
__global__ __launch_bounds__(256)
void k_support_zero(const float* __restrict__ h, const float* __restrict__ h0,
                    float* __restrict__ support, float* __restrict__ agg)
{
    const size_t i = (size_t)blockIdx.x * blockDim.x + threadIdx.x;
    float4 a = ((const float4*)h)[i];
    float4 b = ((const float4*)h0)[i];
    float4 s;
    s.x = (1.f - ALPHA) * a.x + ALPHA * b.x;
    s.y = (1.f - ALPHA) * a.y + ALPHA * b.y;
    s.z = (1.f - ALPHA) * a.z + ALPHA * b.z;
    s.w = (1.f - ALPHA) * a.w + ALPHA * b.w;
    ((float4*)support)[i] = s;
    ((float4*)agg)[i] = make_float4(0.f, 0.f, 0.f, 0.f);
}

// agg[row_e] += w_e * support[col_e]  — scatter-add; gathers hit L2 (51MB << 192MB)
__global__ __launch_bounds__(256)
void k_spmm(const int* __restrict__ erow, const int* __restrict__ ecol,
            const float* __restrict__ ew, const float* __restrict__ support,
            float* __restrict__ agg)
{
    const int e = blockIdx.x * 2 + (threadIdx.x >> 7);
    const int f = threadIdx.x & 127;
    const int r = erow[e];
    const int c = ecol[e];
    const float w = ew[e];
    atomicAdd(&agg[(size_t)r * NHID + f], w * support[(size_t)c * NHID + f]);
}

// h = relu((1-beta)*agg + beta*(support @ Wc))
__global__ __launch_bounds__(256)
void k_combine(const float* __restrict__ support, const float* __restrict__ W,
               const float* __restrict__ agg, float* __restrict__ h)
{
    __shared__ float As[16 * APAD];
    __shared__ float Ws[64 * WPAD];
    const int tid  = threadIdx.x;
    const int wave = tid >> 5;
    const int lane = tid & 31;
    const int half = lane >> 4;
    const int idx  = lane & 15;
    const int rowBase = blockIdx.x * 16;
    const int colBase = wave * 16;
    const int col = colBase + idx;

#pragma unroll
    for (int j = tid; j < 512; j += 256) {
        const int r = j >> 5, c = (j & 31) << 2;
        *(float4*)&As[r * APAD + c] =
            *(const float4*)&support[(size_t)(rowBase + r) * NHID + c];
    }

    v8f acc = {};
    for (int hk = 0; hk < 2; ++hk) {
        const int ks = hk * 64;
        __syncthreads();
#pragma unroll
        for (int j = tid; j < 2048; j += 256) {
            const int r = j >> 5, c = (j & 31) << 2;
            *(float4*)&Ws[r * WPAD + c] =
                *(const float4*)&W[(size_t)(ks + r) * NHID + c];
        }
        __syncthreads();
#pragma unroll
        for (int k = 0; k < 64; k += 4) {
            v2f a, b;
            a.x = As[idx * APAD + ks + k + half * 2 + 0];
            a.y = As[idx * APAD + ks + k + half * 2 + 1];
            b.x = Ws[(k + half * 2 + 0) * WPAD + col];
            b.y = Ws[(k + half * 2 + 1) * WPAD + col];
            acc = __builtin_amdgcn_wmma_f32_16x16x4_f32(false, a, false, b,
                                                        (short)0, acc, false, false);
        }
    }

#pragma unroll
    for (int v = 0; v < 8; ++v) {
        const int row = rowBase + v + half * 8;
        float val = (1.f - BETA) * agg[(size_t)row * NHID + col] + BETA * acc[v];
        val = val > 0.f ? val : 0.f;
        h[(size_t)row * NHID + col] = val;
    }
}

// out = log_softmax(h @ W_out + b_out)  — 4 waves (64 cols), fused softmax via LDS
__global__ __launch_bounds__(128)
void k_out(const float* __restrict__ h, const float* __restrict__ W,
           const float* __restrict__ bias, float* __restrict__ out)
{
    __shared__ float As[16 * APAD];        // 8.4 KB
    __shared__ float Ws[128 * OPAD];       // 36.9 KB (full 128x64 W_out, padded)
    __shared__ float Lg[16 * NCLASS];      // 4 KB logits tile
    const int tid  = threadIdx.x;
    const int wave = tid >> 5;
    const int lane = tid & 31;
    const int half = lane >> 4;
    const int idx  = lane & 15;
    const int rowBase = blockIdx.x * 16;
    const int colBase = wave * 16;
    const int col = colBase + idx;

#pragma unroll
    for (int j = tid; j < 512; j += 128) {
        const int r = j >> 5, c = (j & 31) << 2;
        *(float4*)&As[r * APAD + c] =
            *(const float4*)&h[(size_t)(rowBase + r) * NHID + c];
    }
    // stage W_out: 128 rows x 64 cols = 2048 float4, 16 per thread
#pragma unroll
    for (int j = tid; j < 2048; j += 128) {
        const int r = j >> 4, c = (j & 15) << 2;
        *(float4*)&Ws[r * OPAD + c] =
            *(const float4*)&W[(size_t)r * NCLASS + c];
    }
    __syncthreads();

    v8f acc = {};
#pragma unroll
    for (int k = 0; k < NHID; k += 4) {
        v2f a, b;
        a.x = As[idx * APAD + k + half * 2 + 0];
        a.y = As[idx * APAD + k + half * 2 + 1];
        b.x = Ws[(k + half * 2 + 0) * OPAD + col];
        b.y = Ws[(k + half * 2 + 1) * OPAD + col];
        acc = __builtin_amdgcn_wmma_f32_16x16x4_f32(false, a, false, b,
                                                    (short)0, acc, false, false);
    }

    const float bv = bias[col];
#pragma unroll
    for (int v = 0; v < 8; ++v)
        Lg[(v + half * 8) * NCLASS + col] = acc[v] + bv;
    __syncthreads();

    // log_softmax: one thread per row (16 rows, 64 classes each)
    if (tid < 16) {
        const float* rowp = &Lg[tid * NCLASS];
        float m = rowp[0];
        for (int c = 1; c < NCLASS; ++c) m = fmaxf(m, rowp[c]);
        float s = 0.f;
        for (int c = 0; c < NCLASS; ++c) s += __expf(rowp[c] - m);
        const float lse = m + __logf(s);
        float* op = out + (size_t)(rowBase + tid) * NCLASS;
        for (int c = 0; c < NCLASS; ++c) op[c] = rowp[c] - lse;
    }
}

// ---------------------------------------------------------------------------
extern "C" void kernel_launch(void* const* d_in, const int* in_sizes, int n_in,
                              void* d_out, int out_size, void* d_ws, size_t ws_size,
                              hipStream_t stream)
{
    (void)in_sizes; (void)n_in; (void)out_size; (void)ws_size;

    const float* x     = (const float*)d_in[0];
    const int*   erow  = (const int*)  d_in[1];
    const int*   ecol  = (const int*)  d_in[2];
    const float* ew    = (const float*)d_in[3];
    const float* W_in  = (const float*)d_in[4];
    const float* b_in  = (const float*)d_in[5];
    const float* Wc1   = (const float*)d_in[6];
    const float* Wc2   = (const float*)d_in[7];
    const float* W_out = (const float*)d_in[8];
    const float* b_out = (const float*)d_in[9];
    float* out = (float*)d_out;

    const size_t MATSZ = (size_t)NNODES * NHID;       // 12.8M floats
    float* h0      = (float*)d_ws;
    float* h       = h0 + MATSZ;
    float* support = h  + MATSZ;
    float* agg     = support + MATSZ;

    const int rowTiles  = NNODES / 16;                // 6250
    const int vecBlocks = (int)(MATSZ / 4 / 256);     // 12500

    k_lin_in<<<rowTiles, 256, 0, stream>>>(x, W_in, b_in, h, h0);

    for (int layer = 0; layer < 2; ++layer) {
        const float* Wc = (layer == 0) ? Wc1 : Wc2;
        k_support_zero<<<vecBlocks, 256, 0, stream>>>(h, h0, support, agg);
        k_spmm<<<NEDGES / 2, 256, 0, stream>>>(erow, ecol, ew, support, agg);
        k_combine<<<rowTiles, 256, 0, stream>>>(support, Wc, agg, h);
    }

    k_out<<<rowTiles, 128, 0, stream>>>(h, W_out, b_out, out);
}